// Image_ECF_3D_45457933861132
// MI455X (gfx1250) — compile-verified
//
#include <hip/hip_runtime.h>
#include <cstdint>

// ---------------------------------------------------------------------------
// Image_ECF_3D on MI455X (gfx1250)
//
// Strategy (memory/atomic-bound problem, no GEMM structure):
//  - 256^3 f32 volume. Block = (x, y-chunk) strip, 256 threads spanning z
//    (contiguous axis) => coalesced row loads.
//  - Rows (x,y) / (x+1,y) staged into LDS via gfx1250 async global->LDS
//    copies (ASYNCcnt) with a 3-row ring buffer: prefetch row y+2 while
//    binning voxel slab y.
//  - Per-wave (wave32) private 256-bin LDS histograms -> ds_add atomics with
//    no cross-wave contention; merged once per block with global atomics.
//  - Tiny 256-wide inclusive scan kernel finishes.
// ---------------------------------------------------------------------------

#define ECF_N      256
#define ECF_NBINS  256
#define ROWPAD     264      // 256 + pad so r[z+1] at z==255 stays in-bounds
#define WAVES      8        // 256 threads / wave32
#define YCHUNK     64       // rows of voxels per block

// ---- gfx1250 async global->LDS staging (guarded; falls back to ld+ds_store)
#if defined(__HIP_DEVICE_COMPILE__) && defined(__gfx1250__)
#if __has_builtin(__builtin_amdgcn_global_load_async_to_lds_b32)
#define ECF_HAS_ASYNC_LDS 1
#endif
#endif

typedef __attribute__((address_space(1))) int* glb_i32_ptr;
typedef __attribute__((address_space(3))) int* lds_i32_ptr;

__device__ __forceinline__ void row_load(float* dst_lds, const float* src_glb, int t) {
#if defined(ECF_HAS_ASYNC_LDS)
  __builtin_amdgcn_global_load_async_to_lds_b32(
      (glb_i32_ptr)(uintptr_t)(src_glb + t),
      (lds_i32_ptr)(uint32_t)(uintptr_t)(dst_lds + t),
      0, 0);
#else
  dst_lds[t] = src_glb[t];
#endif
}

__device__ __forceinline__ void row_load_wait() {
#if defined(ECF_HAS_ASYNC_LDS)
#if __has_builtin(__builtin_amdgcn_s_wait_asynccnt)
  __builtin_amdgcn_s_wait_asynccnt(0);
#else
  asm volatile("s_wait_asynccnt 0" ::: "memory");
#endif
#endif
}

__device__ __forceinline__ int bin_of(float v) {
  int b = (int)ceilf(v * 255.0f);        // scale = num_vals - 1
  b = (b < 0) ? 0 : b;
  return (b > 255) ? 255 : b;
}

// ---------------------------------------------------------------------------
__global__ __launch_bounds__(256)
void Image_ECF_3D_hist_kernel(const float* __restrict__ img, int* __restrict__ g_hist) {
  __shared__ float rb[6][ROWPAD];          // [0..2]: rows at x, [3..5]: rows at x+1
  __shared__ int   hist[WAVES][ECF_NBINS]; // per-wave private histograms

  const int  z    = threadIdx.x;           // z coordinate
  const int  wid  = z >> 5;                // wave id (wave32)
  const int  x    = blockIdx.x;            // 0..255
  const int  y0   = blockIdx.y * YCHUNK;
  const int  yend = y0 + YCHUNK;
  const bool hasX = (x < ECF_N - 1);
  const bool hasZ = (z < ECF_N - 1);

#pragma unroll
  for (int w = 0; w < WAVES; ++w) hist[w][z] = 0;

  const size_t base0 = ((((size_t)x)     << 8) + (size_t)y0) << 8;  // (x,   y0, 0)
  const size_t base1 = ((((size_t)x + 1) << 8) + (size_t)y0) << 8;  // (x+1, y0, 0)

  // Preload rows y0 and y0+1 for both x-streams (y0 <= 192 so y0+1 is valid).
  row_load(&rb[0][0], img + base0,       z);
  row_load(&rb[1][0], img + base0 + 256, z);
  if (hasX) {
    row_load(&rb[3][0], img + base1,       z);
    row_load(&rb[4][0], img + base1 + 256, z);
  }
  row_load_wait();
  __syncthreads();

  int i0 = 0, i1 = 1, i2 = 2;              // ring indices: row y, y+1, prefetch
  for (int y = y0; y < yend; ++y) {
    const bool hasY = (y < ECF_N - 1);

    // Prefetch row y+2 into the ring slot freed after iteration y-1.
    if ((y + 1 < yend) && (y + 2 < ECF_N)) {
      const size_t off = (size_t)(y + 2 - y0) << 8;
      row_load(&rb[i2][0], img + base0 + off, z);
      if (hasX) row_load(&rb[3 + i2][0], img + base1 + off, z);
    }

    // 8 cell corners (guards make stale/pad reads harmless).
    const float v000 = rb[i0][z];
    const float v001 = rb[i0][z + 1];
    const float v010 = rb[i1][z];
    const float v011 = rb[i1][z + 1];
    const float v100 = rb[3 + i0][z];
    const float v101 = rb[3 + i0][z + 1];
    const float v110 = rb[3 + i1][z];
    const float v111 = rb[3 + i1][z + 1];

    int* h = hist[wid];
    atomicAdd(&h[bin_of(v000)], 1);                                        // vertex
    if (hasZ) atomicAdd(&h[bin_of(fmaxf(v000, v001))], -1);                // z-edge
    if (hasY) atomicAdd(&h[bin_of(fmaxf(v000, v010))], -1);                // y-edge
    if (hasX) atomicAdd(&h[bin_of(fmaxf(v000, v100))], -1);                // x-edge
    if (hasY && hasZ)                                                      // yz-square
      atomicAdd(&h[bin_of(fmaxf(fmaxf(v000, v001), fmaxf(v010, v011)))], 1);
    if (hasX && hasZ)                                                      // xz-square
      atomicAdd(&h[bin_of(fmaxf(fmaxf(v000, v001), fmaxf(v100, v101)))], 1);
    if (hasX && hasY)                                                      // xy-square
      atomicAdd(&h[bin_of(fmaxf(fmaxf(v000, v010), fmaxf(v100, v110)))], 1);
    if (hasX && hasY && hasZ)                                              // cube
      atomicAdd(&h[bin_of(fmaxf(fmaxf(fmaxf(v000, v001), fmaxf(v010, v011)),
                                fmaxf(fmaxf(v100, v101), fmaxf(v110, v111))))], -1);

    row_load_wait();     // prefetched rows resident for next iteration
    __syncthreads();     // all waves done reading before slots are reused

    const int t = i0; i0 = i1; i1 = i2; i2 = t;
  }

  // Merge the 8 wave-private histograms; one global atomic per (block, bin).
  int sum = 0;
#pragma unroll
  for (int w = 0; w < WAVES; ++w) sum += hist[w][z];
  atomicAdd(&g_hist[z], sum);
}

// ---------------------------------------------------------------------------
__global__ __launch_bounds__(256)
void Image_ECF_3D_zero_kernel(int* __restrict__ g_hist) {
  g_hist[threadIdx.x] = 0;
}

__global__ __launch_bounds__(256)
void Image_ECF_3D_scan_kernel(const int* __restrict__ g_hist, float* __restrict__ out) {
  __shared__ int s[ECF_NBINS];
  const int t = threadIdx.x;
  s[t] = g_hist[t];
  __syncthreads();
#pragma unroll
  for (int off = 1; off < ECF_NBINS; off <<= 1) {
    const int v = (t >= off) ? s[t - off] : 0;
    __syncthreads();
    s[t] += v;
    __syncthreads();
  }
  out[t] = (float)s[t];
}

// ---------------------------------------------------------------------------
extern "C" void kernel_launch(void* const* d_in, const int* in_sizes, int n_in,
                              void* d_out, int out_size, void* d_ws, size_t ws_size,
                              hipStream_t stream) {
  (void)in_sizes; (void)n_in; (void)out_size; (void)ws_size;
  const float* img    = (const float*)d_in[0];
  int*         g_hist = (int*)d_ws;           // 256 ints of scratch
  float*       out    = (float*)d_out;        // 256 outputs

  Image_ECF_3D_zero_kernel<<<dim3(1), dim3(256), 0, stream>>>(g_hist);
  Image_ECF_3D_hist_kernel<<<dim3(ECF_N, ECF_N / YCHUNK), dim3(256), 0, stream>>>(img, g_hist);
  Image_ECF_3D_scan_kernel<<<dim3(1), dim3(256), 0, stream>>>(g_hist, out);
}